// Block_21251498181165
// MI455X (gfx1250) — compile-verified
//
#include <hip/hip_runtime.h>
#include <math.h>

#define N_NODES 20000
#define N_EDGES 320000

typedef __attribute__((ext_vector_type(2))) float v2f;
typedef __attribute__((ext_vector_type(8))) float v8f;

// ---- LDS weight arena offsets (in floats) ----
#define OFF_W000_1 0       // (16,64)
#define OFF_W011_1 1024    // (16,64)
#define OFF_W110_1 2048    // (16,32)
#define OFF_W101_1 2560    // (16,32)
#define OFF_W112_1 3072    // (16,32)
#define OFF_W211_1 3584    // (16,16)
#define OFF_W202_1 3840    // (16,16)
#define OFF_W2     4096    // 7 x (16,8): order [000,110,011,101,211,112,202]
#define OFF_WOA    4992    // (64,32) = [w000_o | w110_o]
#define OFF_WOV    7040    // (32,48) = [w011_o | w101_o | w211_o]
#define OFF_WOD    8576    // (16,32) = [w112_o | w202_o]
#define OFF_ACEA   9088    // (64,8)
#define OFF_ACEV   9600    // (32,8)
#define OFF_ACED   9856    // (16,8)
#define OFF_CBUF   9984    // 8 waves x 16 edges x 48 ranks
#define LDS_FLOATS (OFF_CBUF + 8*16*48)   // 16128 floats = 63 KB

// fp32 WMMA: D(16x16) += A(16x4) * B(4x16), wave32.
__device__ __forceinline__ v8f wmma4(v2f a, v2f b, v8f c) {
  return __builtin_amdgcn_wmma_f32_16x16x4_f32(
      /*neg_a=*/false, a, /*neg_b=*/false, b,
      /*c_mod=*/(short)0, c, /*reuse_a=*/false, /*reuse_b=*/false);
}

// D[rank,edge] = W(16 x 4*nkc) * x^T.  A from LDS weights (row m = lane&15),
// B gathered per-lane from this lane's dst-node feature vector (stride cstep).
__device__ __forceinline__ v8f gemmA(const float* __restrict__ Wl, int m, int ldw,
                                     int nkc, const float* __restrict__ xb,
                                     int cstep, int hk) {
  v8f acc = {0.f,0.f,0.f,0.f,0.f,0.f,0.f,0.f};
  for (int kc = 0; kc < nkc; ++kc) {
    const float* wp = Wl + m*ldw + kc*4 + hk;
    v2f A = { wp[0], wp[1] };
    int k = kc*4 + hk;
    v2f B = { xb[k*cstep], xb[(k+1)*cstep] };
    acc = wmma4(A, B, acc);
  }
  return acc;
}

// Output projection: D[out_ch,edge] = Wo(16 x 4*nkc tile) * c.  B comes from the
// per-wave LDS c-buffer (cb already offset to this lane's edge column).
__device__ __forceinline__ v8f gemmO(const float* __restrict__ Wl, int m, int ldw,
                                     int nkc, const float* __restrict__ cb, int hk) {
  v8f acc = {0.f,0.f,0.f,0.f,0.f,0.f,0.f,0.f};
  for (int kc = 0; kc < nkc; ++kc) {
    const float* wp = Wl + m*ldw + kc*4 + hk;
    v2f A = { wp[0], wp[1] };
    v2f B = { cb[kc*4 + hk], cb[kc*4 + hk + 1] };
    acc = wmma4(A, B, acc);
  }
  return acc;
}

// ACE embed tile: D[ch,edge] = ace_w(16x8 tile) * rad^T (K=8 -> 2 wmma).
__device__ __forceinline__ v8f gemmACE(const float* __restrict__ Wl, int m,
                                       const float rad[8], int hk) {
  v8f acc = {0.f,0.f,0.f,0.f,0.f,0.f,0.f,0.f};
  for (int kc = 0; kc < 2; ++kc) {
    v2f A = { Wl[m*8 + kc*4 + hk], Wl[m*8 + kc*4 + hk + 1] };
    v2f B = { rad[kc*4 + hk], rad[kc*4 + hk + 1] };
    acc = wmma4(A, B, acc);
  }
  return acc;
}

// per-lane b[r] = w2[rbase+j,:] . rad  for this lane's rank half
__device__ __forceinline__ void calcB(const float* __restrict__ w2,
                                      const float rad[8], int rbase, float bo[8]) {
#pragma unroll
  for (int j = 0; j < 8; ++j) {
    const float* wr = w2 + (rbase + j) * 8;
    float a = 0.f;
#pragma unroll
    for (int k = 0; k < 8; ++k) a += wr[k] * rad[k];
    bo[j] = a;
  }
}

__global__ void zero_kernel(float* __restrict__ p, long n) {
  long i = (long)blockIdx.x * blockDim.x + threadIdx.x;
  long stride = (long)gridDim.x * blockDim.x;
  for (; i < n; i += stride) p[i] = 0.f;
}

__global__ __launch_bounds__(256)
void edge_kernel(const float* __restrict__ pos, const float* __restrict__ xA,
                 const float* __restrict__ xV, const float* __restrict__ xD,
                 const float* __restrict__ box,
                 const float* __restrict__ aceA, const float* __restrict__ aceV,
                 const float* __restrict__ aceD,
                 const float* __restrict__ w000_1, const float* __restrict__ w000_2, const float* __restrict__ w000_o,
                 const float* __restrict__ w101_1, const float* __restrict__ w101_2, const float* __restrict__ w101_o,
                 const float* __restrict__ w202_1, const float* __restrict__ w202_2, const float* __restrict__ w202_o,
                 const float* __restrict__ w011_1, const float* __restrict__ w011_2, const float* __restrict__ w011_o,
                 const float* __restrict__ w110_1, const float* __restrict__ w110_2, const float* __restrict__ w110_o,
                 const float* __restrict__ w112_1, const float* __restrict__ w112_2, const float* __restrict__ w112_o,
                 const float* __restrict__ w211_1, const float* __restrict__ w211_2, const float* __restrict__ w211_o,
                 const int* __restrict__ src, const int* __restrict__ dst,
                 float* __restrict__ out)
{
  __shared__ float lds[LDS_FLOATS];
  const int tid = threadIdx.x;

  // ---- stage all weights into LDS (once per block) ----
  for (int i = tid; i < 1024; i += 256) lds[OFF_W000_1 + i] = w000_1[i];
  for (int i = tid; i < 1024; i += 256) lds[OFF_W011_1 + i] = w011_1[i];
  for (int i = tid; i < 512;  i += 256) lds[OFF_W110_1 + i] = w110_1[i];
  for (int i = tid; i < 512;  i += 256) lds[OFF_W101_1 + i] = w101_1[i];
  for (int i = tid; i < 512;  i += 256) lds[OFF_W112_1 + i] = w112_1[i];
  for (int i = tid; i < 256;  i += 256) lds[OFF_W211_1 + i] = w211_1[i];
  for (int i = tid; i < 256;  i += 256) lds[OFF_W202_1 + i] = w202_1[i];
  for (int i = tid; i < 128;  i += 256) {
    lds[OFF_W2 + 0*128 + i] = w000_2[i];
    lds[OFF_W2 + 1*128 + i] = w110_2[i];
    lds[OFF_W2 + 2*128 + i] = w011_2[i];
    lds[OFF_W2 + 3*128 + i] = w101_2[i];
    lds[OFF_W2 + 4*128 + i] = w211_2[i];
    lds[OFF_W2 + 5*128 + i] = w112_2[i];
    lds[OFF_W2 + 6*128 + i] = w202_2[i];
  }
  for (int i = tid; i < 2048; i += 256) {            // (64,32) stacked
    int r = i >> 5, c = i & 31;
    lds[OFF_WOA + i] = (c < 16) ? w000_o[r*16 + c] : w110_o[r*16 + (c-16)];
  }
  for (int i = tid; i < 1536; i += 256) {            // (32,48) stacked
    int r = i / 48, c = i % 48;
    float v;
    if (c < 16)      v = w011_o[r*16 + c];
    else if (c < 32) v = w101_o[r*16 + (c-16)];
    else             v = w211_o[r*16 + (c-32)];
    lds[OFF_WOV + i] = v;
  }
  for (int i = tid; i < 512; i += 256) {             // (16,32) stacked
    int r = i >> 5, c = i & 31;
    lds[OFF_WOD + i] = (c < 16) ? w112_o[r*16 + c] : w202_o[r*16 + (c-16)];
  }
  for (int i = tid; i < 512; i += 256) lds[OFF_ACEA + i] = aceA[i];
  for (int i = tid; i < 256; i += 256) lds[OFF_ACEV + i] = aceV[i];
  for (int i = tid; i < 128; i += 256) lds[OFF_ACED + i] = aceD[i];
  __syncthreads();

  const int wave  = tid >> 5;
  const int lane  = tid & 31;
  const int eloc  = lane & 15;         // edge column / weight row
  const int half  = lane >> 4;
  const int hk    = half << 1;         // K-pair base within each K=4 chunk
  const int rbase = half << 3;         // D row base for this lane half
  const int tile  = blockIdx.x * 8 + wave;
  if (tile * 16 >= N_EDGES) return;    // uniform per wave -> EXEC stays full

  const int e  = tile * 16 + eloc;
  const int sn = src[e];
  const int dn = dst[e];

  // ---- per-edge geometry (duplicated across lane halves) ----
  const float bx0 = box[0], bx1 = box[1], bx2 = box[2];
  float dx = pos[(size_t)dn*3+0] - pos[(size_t)sn*3+0]; dx -= bx0 * rintf(dx/bx0);
  float dy = pos[(size_t)dn*3+1] - pos[(size_t)sn*3+1]; dy -= bx1 * rintf(dy/bx1);
  float dz = pos[(size_t)dn*3+2] - pos[(size_t)sn*3+2]; dz -= bx2 * rintf(dz/bx2);
  const float r2  = dx*dx + dy*dy + dz*dz;
  const float xsq = r2 * 0.25f;                       // |r|^2 / R0, R0 = 4
  const float xx  = sqrtf(xsq + 1e-12f);
  const float env = fmaxf(0.f, 1.f - xsq);
  float rad[8];
#pragma unroll
  for (int k = 0; k < 8; ++k)
    rad[k] = __cosf(3.14159265358979f * (float)k * xx) * env;
  const float gx = dx*4.25f, gy = dy*4.25f, gz = dz*4.25f; // r * 17/R0
  const float gn = sqrtf(gx*gx + gy*gy + gz*gz + 1e-12f);
  const float sg = 1.f / (1.f + __expf(-gn));
  float rh[3] = { gx*sg, gy*sg, gz*sg };

  const float* pA = xA + (size_t)dn * 64;
  const float* pV = xV + (size_t)dn * 96;
  const float* pD = xD + (size_t)dn * 144;

  float* cb   = lds + OFF_CBUF + wave*768 + eloc*48;  // this lane's c column
  float* outA = out;
  float* outV = out + (size_t)N_NODES * 64;
  float* outD = out + (size_t)N_NODES * 160;          // 64 + 96

  // ================= psi_a : 00->0 (xa,rad) + 11->0 (xv,phi_v) =================
  {
    v8f a000 = gemmA(lds + OFF_W000_1, eloc, 64, 16, pA, 1, hk);
    float b0[8]; calcB(lds + OFF_W2 + 0*128, rad, rbase, b0);
    float b1[8]; calcB(lds + OFF_W2 + 1*128, rad, rbase, b1);
    v8f t = {0.f,0.f,0.f,0.f,0.f,0.f,0.f,0.f};
#pragma unroll
    for (int s = 0; s < 3; ++s) {
      v8f av = gemmA(lds + OFF_W110_1, eloc, 32, 8, pV + s, 3, hk);
#pragma unroll
      for (int j = 0; j < 8; ++j) t[j] += av[j] * rh[s];
    }
#pragma unroll
    for (int j = 0; j < 8; ++j) {
      cb[rbase + j]      = a000[j] * b0[j];
      cb[16 + rbase + j] = t[j]    * b1[j];
    }
    asm volatile("" ::: "memory");   // DS in-order per wave; block compile-time reordering
#pragma unroll
    for (int mt = 0; mt < 4; ++mt) {
      v8f o  = gemmO(lds + OFF_WOA, mt*16 + eloc, 32, 8, cb, hk);
      v8f ba = gemmACE(lds + OFF_ACEA, mt*16 + eloc, rad, hk);
      float* op = outA + (size_t)sn * 64 + mt*16 + rbase;
#pragma unroll
      for (int j = 0; j < 8; ++j) atomicAdd(&op[j], o[j] + ba[j]);
    }
    asm volatile("" ::: "memory");
  }

  // ========= psi_v : 01->1 (xa) + 10->1 (xv) + 21->1 (xd), per spatial i =========
  {
    v8f a011 = gemmA(lds + OFF_W011_1, eloc, 64, 16, pA, 1, hk);
    float b2[8]; calcB(lds + OFF_W2 + 2*128, rad, rbase, b2);
    float b3[8]; calcB(lds + OFF_W2 + 3*128, rad, rbase, b3);
    float b4[8]; calcB(lds + OFF_W2 + 4*128, rad, rbase, b4);
    v8f bv0 = gemmACE(lds + OFF_ACEV, eloc,      rad, hk);
    v8f bv1 = gemmACE(lds + OFF_ACEV, 16 + eloc, rad, hk);
#pragma unroll
    for (int i = 0; i < 3; ++i) {
      v8f a101 = gemmA(lds + OFF_W101_1, eloc, 32, 8, pV + i, 3, hk);
      v8f t = {0.f,0.f,0.f,0.f,0.f,0.f,0.f,0.f};
#pragma unroll
      for (int jj = 0; jj < 3; ++jj) {
        v8f ad = gemmA(lds + OFF_W211_1, eloc, 16, 4, pD + i*3 + jj, 9, hk);
#pragma unroll
        for (int j = 0; j < 8; ++j) t[j] += ad[j] * rh[jj];
      }
#pragma unroll
      for (int j = 0; j < 8; ++j) {
        cb[rbase + j]      = a011[j] * b2[j] * rh[i];
        cb[16 + rbase + j] = a101[j] * b3[j];
        cb[32 + rbase + j] = t[j]    * b4[j];
      }
      asm volatile("" ::: "memory");
#pragma unroll
      for (int mt = 0; mt < 2; ++mt) {
        v8f o = gemmO(lds + OFF_WOV, mt*16 + eloc, 48, 12, cb, hk);
        v8f bv = mt ? bv1 : bv0;
        float* op = outV + (size_t)sn * 96;
#pragma unroll
        for (int j = 0; j < 8; ++j)
          atomicAdd(&op[(mt*16 + rbase + j)*3 + i], o[j] + bv[j]*rh[i]);
      }
      asm volatile("" ::: "memory");
    }
  }

  // ========= psi_d : 11->2 (xv) + 20->2 (xd), per spatial (i,jj) =========
  {
    float b5[8]; calcB(lds + OFF_W2 + 5*128, rad, rbase, b5);
    float b6[8]; calcB(lds + OFF_W2 + 6*128, rad, rbase, b6);
    v8f bd = gemmACE(lds + OFF_ACED, eloc, rad, hk);
#pragma unroll
    for (int i = 0; i < 3; ++i) {
      v8f a112 = gemmA(lds + OFF_W112_1, eloc, 32, 8, pV + i, 3, hk);
#pragma unroll
      for (int jj = 0; jj < 3; ++jj) {
        v8f a202 = gemmA(lds + OFF_W202_1, eloc, 16, 4, pD + i*3 + jj, 9, hk);
#pragma unroll
        for (int j = 0; j < 8; ++j) {
          cb[rbase + j]      = a112[j] * b5[j] * rh[jj];
          cb[16 + rbase + j] = a202[j] * b6[j];
        }
        asm volatile("" ::: "memory");
        v8f o = gemmO(lds + OFF_WOD, eloc, 32, 8, cb, hk);
        const float rr = rh[i] * rh[jj];
        float* op = outD + (size_t)sn * 144;
#pragma unroll
        for (int j = 0; j < 8; ++j)
          atomicAdd(&op[(rbase + j)*9 + i*3 + jj], o[j] + bd[j]*rr);
        asm volatile("" ::: "memory");
      }
    }
  }
}

extern "C" void kernel_launch(void* const* d_in, const int* in_sizes, int n_in,
                              void* d_out, int out_size, void* d_ws, size_t ws_size,
                              hipStream_t stream)
{
  const float* pos    = (const float*)d_in[0];
  const float* x_a    = (const float*)d_in[1];
  const float* x_v    = (const float*)d_in[2];
  const float* x_d    = (const float*)d_in[3];
  const float* box    = (const float*)d_in[4];
  const float* ace_wa = (const float*)d_in[5];
  const float* ace_wv = (const float*)d_in[6];
  const float* ace_wd = (const float*)d_in[7];
  const float* w000_1 = (const float*)d_in[8];
  const float* w000_2 = (const float*)d_in[9];
  const float* w000_o = (const float*)d_in[10];
  const float* w101_1 = (const float*)d_in[11];
  const float* w101_2 = (const float*)d_in[12];
  const float* w101_o = (const float*)d_in[13];
  const float* w202_1 = (const float*)d_in[14];
  const float* w202_2 = (const float*)d_in[15];
  const float* w202_o = (const float*)d_in[16];
  const float* w011_1 = (const float*)d_in[17];
  const float* w011_2 = (const float*)d_in[18];
  const float* w011_o = (const float*)d_in[19];
  const float* w110_1 = (const float*)d_in[20];
  const float* w110_2 = (const float*)d_in[21];
  const float* w110_o = (const float*)d_in[22];
  const float* w112_1 = (const float*)d_in[23];
  const float* w112_2 = (const float*)d_in[24];
  const float* w112_o = (const float*)d_in[25];
  const float* w211_1 = (const float*)d_in[26];
  const float* w211_2 = (const float*)d_in[27];
  const float* w211_o = (const float*)d_in[28];
  const int*   src    = (const int*)d_in[29];
  const int*   dst    = (const int*)d_in[30];
  float* out = (float*)d_out;

  // zero the accumulation output (harness poisons it with 0xAA)
  zero_kernel<<<2048, 256, 0, stream>>>(out, (long)out_size);

  // 8 waves/block * 16 edges/wave = 128 edges per block; 2500 blocks = 320000 edges
  edge_kernel<<<N_EDGES / 128, 256, 0, stream>>>(
      pos, x_a, x_v, x_d, box, ace_wa, ace_wv, ace_wd,
      w000_1, w000_2, w000_o, w101_1, w101_2, w101_o, w202_1, w202_2, w202_o,
      w011_1, w011_2, w011_o, w110_1, w110_2, w110_o, w112_1, w112_2, w112_o,
      w211_1, w211_2, w211_o, src, dst, out);
}